// ODEBlock_48842368090278
// MI455X (gfx1250) — compile-verified
//
#include <hip/hip_runtime.h>
#include <math.h>

// Problem dims
#define BROWS 16384
#define DDIM  256
#define BD    (BROWS * DDIM)      // 4,194,304 elements
#define NPART 4096                // finalize partial-sum blocks

#define WSTRIDE 544               // k-pair interleaved W tile row stride (floats)

typedef float v2f __attribute__((ext_vector_type(2)));
typedef float v8f __attribute__((ext_vector_type(8)));

struct Scalars { float t; float dt; float h; int done; int accept; };
struct Coeffs  { float a[6]; int nk; };

// ---------------------------------------------------------------- init
__global__ void init_kernel(const float* __restrict__ x, float* __restrict__ y,
                            Scalars* sc) {
  int i = blockIdx.x * blockDim.x + threadIdx.x;     // one float4 each
  ((float4*)y)[i] = ((const float4*)x)[i];
  if (blockIdx.x == 0 && threadIdx.x == 0) {
    sc->t = 0.0f; sc->dt = 0.05f; sc->h = 0.0f; sc->done = 0; sc->accept = 0;
  }
}

// ---------------------------------------------------------------- prep (per RK step)
__global__ void prep_kernel(Scalars* sc) {
  float t = sc->t, dt = sc->dt;
  sc->done = (t >= 1.0f - 1e-7f) ? 1 : 0;
  sc->h = fminf(dt, 1.0f - t);
}

// ---------------------------------------------------------------- fused stage GEMM
// k_out = tanh( (y + h * sum_j a[j]*k_j) @ W + bias )
// Block: 256 threads (8 waves). Output tile: 64 (M) x 256 (N). K chunked by 32.
__global__ __launch_bounds__(256)
void stage_kernel(const float* __restrict__ y, const float* __restrict__ kbuf,
                  const float* __restrict__ W, const float* __restrict__ bias,
                  float* __restrict__ kout, const Scalars* __restrict__ sc,
                  Coeffs co) {
  // LDS: A tile 64x32 (row stride 36 -> conflict-free, 16B-aligned rows)
  //      W tile, k-pair interleaved: Ws[k>>1][n][2], row stride 544 floats
  //      -> B fragment (W[k][n], W[k+1][n]) is one aligned b64, conflict-free
  __shared__ __align__(16) float As[64 * 36];
  __shared__ __align__(16) float Ws[16 * WSTRIDE];

  const float h = sc->h;
  const int mtile = blockIdx.x * 64;

  const int lane = threadIdx.x & 31;
  const int wave = threadIdx.x >> 5;
  const int wm   = (wave >> 2) * 32;       // 0 or 32   (M offset within tile)
  const int wn   = (wave & 3) * 64;        // 0/64/128/192 (N offset)
  const int half = lane >> 4;              // 0: lanes 0-15, 1: lanes 16-31
  const int l15  = lane & 15;

  v8f acc[2][4];
#pragma unroll
  for (int ms = 0; ms < 2; ++ms)
#pragma unroll
    for (int ns = 0; ns < 4; ++ns)
      acc[ms][ns] = (v8f){0.f,0.f,0.f,0.f,0.f,0.f,0.f,0.f};

  for (int kb = 0; kb < DDIM; kb += 32) {
    __syncthreads();
    // ---- stage A tile: yi = y + h * sum a[j] * k_j  (coalesced float4)
    {
      const int r0 = threadIdx.x >> 3;           // 0..31
      const int c0 = (threadIdx.x & 7) * 4;      // 0..28
#pragma unroll
      for (int it = 0; it < 2; ++it) {
        const int r = r0 + it * 32;
        const size_t goff = (size_t)(mtile + r) * DDIM + kb + c0;
        float4 v = *(const float4*)(y + goff);
        for (int j = 0; j < co.nk; ++j) {
          const float4 kv = *(const float4*)(kbuf + (size_t)j * BD + goff);
          const float ha = h * co.a[j];
          v.x = fmaf(ha, kv.x, v.x); v.y = fmaf(ha, kv.y, v.y);
          v.z = fmaf(ha, kv.z, v.z); v.w = fmaf(ha, kv.w, v.w);
        }
        *(float4*)&As[r * 36 + c0] = v;          // 144*r + 16*c' : 16B aligned
      }
    }
    // ---- stage W tile 32x256, interleaving adjacent k rows
    {
      const int kp = threadIdx.x >> 4;           // k-pair 0..15
#pragma unroll
      for (int it = 0; it < 4; ++it) {
        const int c = (threadIdx.x & 15) * 4 + it * 64;   // 0..252, mult of 4
        const float4 w0 = *(const float4*)(W + (size_t)(kb + 2 * kp) * DDIM + c);
        const float4 w1 = *(const float4*)(W + (size_t)(kb + 2 * kp + 1) * DDIM + c);
        const int base = kp * WSTRIDE + 2 * c;            // 16B aligned
        *(float4*)&Ws[base]     = make_float4(w0.x, w1.x, w0.y, w1.y);
        *(float4*)&Ws[base + 4] = make_float4(w0.z, w1.z, w0.w, w1.w);
      }
    }
    __syncthreads();

    // ---- 8 WMMA K-steps of 4 over this 32-chunk
#pragma unroll
    for (int kk = 0; kk < 32; kk += 4) {
      const int krow  = kk + 2 * half;           // lanes 0-15: K=kk,kk+1 ; 16-31: kk+2,kk+3
      const int kpair = (kk >> 1) + half;        // Ws row holding (K=krow, K=krow+1)
      v2f afrag[2];
#pragma unroll
      for (int ms = 0; ms < 2; ++ms)
        afrag[ms] = *(const v2f*)&As[(wm + ms * 16 + l15) * 36 + krow];
      v2f bfrag[4];
#pragma unroll
      for (int ns = 0; ns < 4; ++ns)
        bfrag[ns] = *(const v2f*)&Ws[kpair * WSTRIDE + 2 * (wn + ns * 16 + l15)];
#pragma unroll
      for (int ms = 0; ms < 2; ++ms)
#pragma unroll
        for (int ns = 0; ns < 4; ++ns)
          acc[ms][ns] = __builtin_amdgcn_wmma_f32_16x16x4_f32(
              false, afrag[ms], false, bfrag[ns], (short)0, acc[ms][ns],
              false, false);
    }
  }

  // ---- epilogue: tanh(acc + bias) -> k_out
#pragma unroll
  for (int ms = 0; ms < 2; ++ms)
#pragma unroll
    for (int ns = 0; ns < 4; ++ns) {
      const int col = wn + ns * 16 + l15;
      const float bv = bias[col];
#pragma unroll
      for (int r = 0; r < 8; ++r) {
        const int row = mtile + wm + ms * 16 + half * 8 + r;
        kout[(size_t)row * DDIM + col] = tanhf(acc[ms][ns][r] + bv);
      }
    }
}

// ---------------------------------------------------------------- finalize
// y5 = y + h*sum b5_i k_i ; err = h*sum be_i k_i ; partial sums of (err/scale)^2
__global__ __launch_bounds__(256)
void finalize_kernel(const float* __restrict__ y, const float* __restrict__ kbuf,
                     float* __restrict__ ynew, const Scalars* __restrict__ sc,
                     float* __restrict__ partials) {
  const float B5[7] = {(float)(35.0/384.0), 0.0f, (float)(500.0/1113.0),
                       (float)(125.0/192.0), (float)(-2187.0/6784.0),
                       (float)(11.0/84.0), 0.0f};
  const float BE[7] = {(float)(35.0/384.0 - 5179.0/57600.0), 0.0f,
                       (float)(500.0/1113.0 - 7571.0/16695.0),
                       (float)(125.0/192.0 - 393.0/640.0),
                       (float)(-2187.0/6784.0 + 92097.0/339200.0),
                       (float)(11.0/84.0 - 187.0/2100.0),
                       (float)(-1.0/40.0)};
  const float h = sc->h;
  const int i = blockIdx.x * 256 + threadIdx.x;      // one float4 each
  const float4 yv = ((const float4*)y)[i];
  float4 y5 = yv;
  float4 er = make_float4(0.f, 0.f, 0.f, 0.f);
#pragma unroll
  for (int j = 0; j < 7; ++j) {
    const float4 kv = ((const float4*)kbuf)[(size_t)j * (BD / 4) + i];
    const float hb = h * B5[j], he = h * BE[j];
    y5.x = fmaf(hb, kv.x, y5.x); er.x = fmaf(he, kv.x, er.x);
    y5.y = fmaf(hb, kv.y, y5.y); er.y = fmaf(he, kv.y, er.y);
    y5.z = fmaf(hb, kv.z, y5.z); er.z = fmaf(he, kv.z, er.z);
    y5.w = fmaf(hb, kv.w, y5.w); er.w = fmaf(he, kv.w, er.w);
  }
  ((float4*)ynew)[i] = y5;

  float local = 0.f;
  {
    float s, e;
    s = 1e-7f + 1e-5f * fmaxf(fabsf(yv.x), fabsf(y5.x)); e = er.x / s; local += e * e;
    s = 1e-7f + 1e-5f * fmaxf(fabsf(yv.y), fabsf(y5.y)); e = er.y / s; local += e * e;
    s = 1e-7f + 1e-5f * fmaxf(fabsf(yv.z), fabsf(y5.z)); e = er.z / s; local += e * e;
    s = 1e-7f + 1e-5f * fmaxf(fabsf(yv.w), fabsf(y5.w)); e = er.w / s; local += e * e;
  }
  __shared__ float red[256];
  red[threadIdx.x] = local;
  __syncthreads();
  for (int s = 128; s > 0; s >>= 1) {
    if (threadIdx.x < s) red[threadIdx.x] += red[threadIdx.x + s];
    __syncthreads();
  }
  if (threadIdx.x == 0) partials[blockIdx.x] = red[0];   // no atomics -> deterministic
}

// ---------------------------------------------------------------- control (1 block)
__global__ __launch_bounds__(256)
void control_kernel(Scalars* sc, const float* __restrict__ partials) {
  __shared__ float red[256];
  float s = 0.f;
  for (int i = threadIdx.x; i < NPART; i += 256) s += partials[i];  // fixed order
  red[threadIdx.x] = s;
  __syncthreads();
  for (int w = 128; w > 0; w >>= 1) {
    if (threadIdx.x < w) red[threadIdx.x] += red[threadIdx.x + w];
    __syncthreads();
  }
  if (threadIdx.x == 0) {
    const float h = sc->h;
    const int done = sc->done;
    const float err_norm = sqrtf(red[0] / (float)BD);
    const int accept = (err_norm <= 1.0f) && !done;
    float factor = 0.9f * powf(fmaxf(err_norm, 1e-10f), -0.2f);
    factor = fminf(fmaxf(factor, 0.2f), 10.0f);
    if (accept) sc->t = sc->t + h;
    if (!done) sc->dt = h * factor;
    sc->accept = accept;
  }
}

// ---------------------------------------------------------------- commit / copy
__global__ void commit_kernel(float* __restrict__ y, const float* __restrict__ ynew,
                              const Scalars* __restrict__ sc) {
  if (!sc->accept) return;                         // uniform branch
  const int i = blockIdx.x * blockDim.x + threadIdx.x;
  ((float4*)y)[i] = ((const float4*)ynew)[i];
}

__global__ void copyout_kernel(const float* __restrict__ y, float* __restrict__ out) {
  const int i = blockIdx.x * blockDim.x + threadIdx.x;
  ((float4*)out)[i] = ((const float4*)y)[i];
}

// ---------------------------------------------------------------- launch
extern "C" void kernel_launch(void* const* d_in, const int* in_sizes, int n_in,
                              void* d_out, int out_size, void* d_ws, size_t ws_size,
                              hipStream_t stream) {
  const float* x    = (const float*)d_in[0];
  const float* W    = (const float*)d_in[1];
  const float* bias = (const float*)d_in[2];
  float* out = (float*)d_out;

  float* y        = (float*)d_ws;             // BD
  float* ynew     = y + BD;                   // BD
  float* kbuf     = ynew + BD;                // 7 * BD
  float* partials = kbuf + (size_t)7 * BD;    // NPART
  Scalars* sc     = (Scalars*)(partials + NPART);

  static const float A_TAB[7][6] = {
    {0.f, 0.f, 0.f, 0.f, 0.f, 0.f},
    {0.2f, 0.f, 0.f, 0.f, 0.f, 0.f},
    {(float)(3.0/40.0), (float)(9.0/40.0), 0.f, 0.f, 0.f, 0.f},
    {(float)(44.0/45.0), (float)(-56.0/15.0), (float)(32.0/9.0), 0.f, 0.f, 0.f},
    {(float)(19372.0/6561.0), (float)(-25360.0/2187.0), (float)(64448.0/6561.0),
     (float)(-212.0/729.0), 0.f, 0.f},
    {(float)(9017.0/3168.0), (float)(-355.0/33.0), (float)(46732.0/5247.0),
     (float)(49.0/176.0), (float)(-5103.0/18656.0), 0.f},
    {(float)(35.0/384.0), 0.f, (float)(500.0/1113.0), (float)(125.0/192.0),
     (float)(-2187.0/6784.0), (float)(11.0/84.0)},
  };

  init_kernel<<<4096, 256, 0, stream>>>(x, y, sc);
  for (int s = 0; s < 64; ++s) {
    prep_kernel<<<1, 1, 0, stream>>>(sc);
    for (int i = 0; i < 7; ++i) {
      Coeffs co;
      co.nk = i;
      for (int j = 0; j < 6; ++j) co.a[j] = A_TAB[i][j];
      stage_kernel<<<BROWS / 64, 256, 0, stream>>>(y, kbuf, W, bias,
                                                   kbuf + (size_t)i * BD, sc, co);
    }
    finalize_kernel<<<NPART, 256, 0, stream>>>(y, kbuf, ynew, sc, partials);
    control_kernel<<<1, 256, 0, stream>>>(sc, partials);
    commit_kernel<<<4096, 256, 0, stream>>>(y, ynew, sc);
  }
  copyout_kernel<<<4096, 256, 0, stream>>>(y, out);
}